// HornSchunck_64888365908122
// MI455X (gfx1250) — compile-verified
//
#include <hip/hip_runtime.h>
#include <stdint.h>

// Problem constants (match reference)
#define BATCH   8
#define HH      2000
#define WW      2000
#define K_ITERS 5
#define ALPHA_C 10.0f
#define POOL    10
#define PW      (WW / POOL)      // 200 pooled cols/rows

// Tiling
#define TILE    40               // output tile edge (pool aligned)
#define HALO    6                // 5 iteration rings + 1 for derivative stencil
#define TP      (TILE + 2*HALO)  // 52: LDS window edge
#define NT      (TP * TP)        // 2704 cells
#define NTHREADS 256

typedef unsigned int u32x4 __attribute__((ext_vector_type(4)));
typedef int          i32x4 __attribute__((ext_vector_type(4)));
typedef int          i32x8 __attribute__((ext_vector_type(8)));
typedef float        f32x2 __attribute__((ext_vector_type(2)));

#if defined(__HIP_DEVICE_COMPILE__) && __has_builtin(__builtin_amdgcn_tensor_load_to_lds)
#define HAVE_TDM 1
#else
#define HAVE_TDM 0
#endif

// Issue one TDM 2D tile load: 52x52 fp32 tile starting at (gy0,gx0) of a
// HHxWW fp32 tensor -> contiguous LDS (row stride 52 dwords).
__device__ __forceinline__ void tdm_tile_load(const float* __restrict__ gsrc,
                                              int gy0, int gx0,
                                              float* lds_dst) {
#if HAVE_TDM
    uint64_t gaddr = (uint64_t)(uintptr_t)gsrc +
                     4ull * ((uint64_t)gy0 * (uint64_t)WW + (uint64_t)gx0);
    // ---- D# group 0 (128b): count=1 | lds_addr | global_addr[56:0] | type=2
    u32x4 g0;
    g0[0] = 1u;                                        // count=1, user descriptor
    g0[1] = (unsigned)(uintptr_t)lds_dst;              // LDS byte address
    g0[2] = (unsigned)gaddr;                           // global_addr[31:0]
    g0[3] = ((unsigned)(gaddr >> 32) & 0x01FFFFFFu)    // global_addr[56:32]
            | (2u << 30);                              // type = 2 ("image")
    // ---- D# group 1 (256b)
    unsigned td0 = (unsigned)(WW - gx0);               // valid extent from tile start
    unsigned td1 = (unsigned)(HH - gy0);
    i32x8 g1;
    g1[0] = (int)(2u << 16);                           // data_size=4B; mask/iter/pad = 0
    g1[1] = (int)((td0 & 0xFFFFu) << 16);              // tensor_dim0[15:0] (bits 63:48)
    g1[2] = (int)((td0 >> 16) | ((td1 & 0xFFFFu) << 16));   // dim0 hi | dim1 lo
    g1[3] = (int)((td1 >> 16) | ((unsigned)TP << 16)); // dim1 hi | tile_dim0 = 52
    g1[4] = (int)TP;                                   // tile_dim1 = 52, tile_dim2 = 0
    g1[5] = (int)WW;                                   // tensor_dim0_stride[31:0] = 2000
    g1[6] = 0;                                         // stride hi | dim1_stride lo
    g1[7] = 0;
    i32x4 z4 = {0, 0, 0, 0};
#if defined(__clang_major__) && (__clang_major__ >= 23)
    i32x8 z8 = {0, 0, 0, 0, 0, 0, 0, 0};
    __builtin_amdgcn_tensor_load_to_lds(g0, g1, z4, z4, z8, 0);
#else
    __builtin_amdgcn_tensor_load_to_lds(g0, g1, z4, z4, 0);
#endif
#else
    (void)gsrc; (void)gy0; (void)gx0; (void)lds_dst;
#endif
}

__global__ __launch_bounds__(NTHREADS)
void HornSchunck_64888365908122_kernel(const float* __restrict__ x,
                                       const float* __restrict__ y,
                                       float* __restrict__ out) {
    // LDS layout (floats): 8*NT total = 86.5 KB
    //   [0*NT..2*NT)  sV   : packed (vx, vy)   ping
    //   [2*NT..4*NT)  sVB  : packed pong; init staging: x at 2*NT, y at 3*NT
    //   [4*NT..6*NT)  sG   : packed (Ix, Iy)
    //   [6*NT..8*NT)  sC   : packed (It, 1/Ixy)
    extern __shared__ float smem[];
    f32x2* sV  = (f32x2*)(smem + 0 * NT);
    f32x2* sVB = (f32x2*)(smem + 2 * NT);
    f32x2* sG  = (f32x2*)(smem + 4 * NT);
    f32x2* sC  = (f32x2*)(smem + 6 * NT);
    float* stX = smem + 2 * NT;     // staging (aliases sVB; dead after init)
    float* stY = smem + 3 * NT;

    const int bx = blockIdx.x, by = blockIdx.y, b = blockIdx.z;
    // Clamp window origin into the image; clamping LDS indices to the window
    // then equals replicate padding at true image edges, and only corrupts
    // dead halo rings elsewhere (shrinking-validity scheme).
    int gx0 = bx * TILE - HALO;
    int gy0 = by * TILE - HALO;
    gx0 = gx0 < 0 ? 0 : (gx0 > WW - TP ? WW - TP : gx0);
    gy0 = gy0 < 0 ? 0 : (gy0 > HH - TP ? HH - TP : gy0);

    const float* xb = x + (size_t)b * HH * WW;
    const float* yb = y + (size_t)b * HH * WW;
    const int tid = threadIdx.x;

    // ---------------- Stage x,y tiles into LDS via the Tensor Data Mover ------
#if HAVE_TDM
    if (tid < 32) {                 // wave 0 issues both DMAs
        tdm_tile_load(xb, gy0, gx0, stX);
        tdm_tile_load(yb, gy0, gx0, stY);
        __builtin_amdgcn_s_wait_tensorcnt(0);
    }
#else
    for (int i = tid; i < NT; i += NTHREADS) {
        int lr = i / TP, lc = i - lr * TP;
        size_t g = (size_t)(gy0 + lr) * WW + (size_t)(gx0 + lc);
        stX[i] = xb[g];
        stY[i] = yb[g];
    }
#endif
    __syncthreads();

    // ---------------- Init: (vx,vy), (Ix,Iy), (It, 1/Ixy) ---------------------
    for (int i = tid; i < NT; i += NTHREADS) {
        int lr = i / TP, lc = i - lr * TP;
        int gr = gy0 + lr, gc = gx0 + lc;
        int il = lr * TP + (lc > 0      ? lc - 1 : 0);
        int ir = lr * TP + (lc < TP - 1 ? lc + 1 : TP - 1);
        int iu = (lr > 0      ? lr - 1 : 0)      * TP + lc;
        int id = (lr < TP - 1 ? lr + 1 : TP - 1) * TP + lc;

        float xc = stX[i];
        float yc = stY[i];
        float xm = (gc > 0)      ? stX[il] : 0.0f;   // zero-padded central diffs
        float xp = (gc < WW - 1) ? stX[ir] : 0.0f;
        float yu = (gr > 0)      ? stY[iu] : 0.0f;
        float yd = (gr < HH - 1) ? stY[id] : 0.0f;
        float yl = (gc > 0)      ? stY[il] : 0.0f;

        float ix = xp - xm;
        float iy = yd - yu;
        f32x2 v;  v.x = xc - yl;                 // x - y shifted right (0 left col)
                  v.y = xc - yu;                 // x - y shifted down  (0 top row)
        f32x2 g;  g.x = ix; g.y = iy;
        f32x2 c;  c.x = yc - xc;                 // It
                  c.y = 1.0f / (ALPHA_C + ix * ix + iy * iy);
        sV[i] = v;
        sG[i] = g;
        sC[i] = c;
    }
    __syncthreads();

    // ---------------- 5 fused Horn-Schunck iterations, packed f32x2 -----------
    f32x2* vA = sV;
    f32x2* vB = sVB;    // overwrites dead x/y staging on iteration 1
    #pragma unroll 1
    for (int k = 0; k < K_ITERS; ++k) {
        for (int i = tid; i < NT; i += NTHREADS) {
            int lr = i / TP, lc = i - lr * TP;
            int il = lr * TP + (lc > 0      ? lc - 1 : 0);
            int ir = lr * TP + (lc < TP - 1 ? lc + 1 : TP - 1);
            int iu = (lr > 0      ? lr - 1 : 0)      * TP + lc;
            int id = (lr < TP - 1 ? lr + 1 : TP - 1) * TP + lc;

            f32x2 c0 = vA[i];
            f32x2 a = c0 - vA[il];        // backward x-diff (replicate pad)
            f32x2 r = vA[ir] - c0;        // forward  x-diff
            f32x2 u = c0 - vA[iu];        // backward y-diff
            f32x2 d = vA[id] - c0;        // forward  y-diff
            f32x2 vb = 0.25f * (a * a + r * r + u * u + d * d);

            f32x2 g = sG[i];              // (Ix, Iy)
            f32x2 c = sC[i];              // (It, 1/Ixy)
            float coeff = (g.x * vb.x + g.y * vb.y + c.x) * c.y;
            vB[i] = vb - coeff * g;
        }
        __syncthreads();
        f32x2* t = vA; vA = vB; vB = t;
    }
    // Final (vx, vy) live packed in vA.

    // ---------------- 10x10 max-pool, all 256 lanes + shfl reduction -----------
    {
        int pair = tid >> 3;           // 32 (channel, window) pairs
        int sub  = tid & 7;            // 8 lanes cooperate per pair
        int ch   = pair >> 4;          // 0 -> vx, 1 -> vy
        int wid  = pair & 15;          // 16 windows per tile
        int wy = wid >> 2, wx = wid & 3;
        int offr = by * TILE - gy0;    // output-region offset inside window
        int offc = bx * TILE - gx0;    // (== HALO for interior tiles)
        int r0 = offr + wy * POOL;
        int c0 = offc + wx * POOL;
        float m = -3.402823466e38f;
        for (int e = sub; e < POOL * POOL; e += 8) {
            int dr = e / POOL, dc = e - dr * POOL;
            f32x2 v2 = vA[(r0 + dr) * TP + (c0 + dc)];
            float val = ch ? v2.y : v2.x;
            m = val > m ? val : m;
        }
        // reduce 8 lanes (aligned subgroups of the wave32)
        for (int s = 1; s < 8; s <<= 1) {
            float o = __shfl_xor(m, s, 32);
            m = o > m ? o : m;
        }
        if (sub == 0) {
            int pr = by * (TILE / POOL) + wy;
            int pc = bx * (TILE / POOL) + wx;
            out[(((size_t)b * 2 + ch) * PW + pr) * PW + pc] = m;
        }
    }
}

extern "C" void kernel_launch(void* const* d_in, const int* in_sizes, int n_in,
                              void* d_out, int out_size, void* d_ws, size_t ws_size,
                              hipStream_t stream) {
    (void)in_sizes; (void)n_in; (void)d_ws; (void)ws_size; (void)out_size;
    const float* x = (const float*)d_in[0];
    const float* y = (const float*)d_in[1];
    float* out = (float*)d_out;
    dim3 grid(WW / TILE, HH / TILE, BATCH);            // 50 x 50 x 8
    size_t shmem = (size_t)8 * NT * sizeof(float);     // 86.5 KB dynamic LDS
    HornSchunck_64888365908122_kernel<<<grid, NTHREADS, shmem, stream>>>(x, y, out);
}